// KNNSpaceMean_62088047231498
// MI455X (gfx1250) — compile-verified
//
#include <hip/hip_runtime.h>

typedef __attribute__((ext_vector_type(2))) float v2f;
typedef __attribute__((ext_vector_type(8))) float v8f;

#define B_   4
#define N_   4096
#define C_   64
#define MAXK 16
#define QPB  128   // queries per block
#define TPB  256   // 8 waves (wave32)

__global__ __launch_bounds__(TPB)
void knn_space_mean_kernel(const float* __restrict__ points,
                           const float* __restrict__ preds,
                           const float* __restrict__ kvec,
                           float* __restrict__ out)
{
    __shared__ int s_idx[QPB][MAXK];

    const int tid  = threadIdx.x;
    const int lane = tid & 31;
    const int wave = tid >> 5;
    const int half = lane >> 4;   // which K-half of the WMMA operands
    const int col  = lane & 15;

    const int bid   = blockIdx.x;
    const int b     = bid / (N_ / QPB);
    const int qbase = (bid % (N_ / QPB)) * QPB;

    const float* ptsB = points + (size_t)b * N_ * 3;

    // ---- k = argmax(k_vector) + 1 (softmax is monotonic; first-max tiebreak) ----
    int kk = 1;
    {
        float best = kvec[0];
        #pragma unroll
        for (int j = 1; j < MAXK; ++j) {
            float v = kvec[j];
            if (v > best) { best = v; kk = j + 1; }
        }
    }

    // ---- Phase 1: fused score tiles via WMMA + in-register top-16 ----
    // Wave owns 16 queries. Query n is shared by lane pair (n, n+16):
    // lane half h scans candidate rows m = 8h..8h+7 of every 16-wide tile.
    const int gq = qbase + wave * 16 + col;   // this lane's query

    const float qx = ptsB[gq * 3 + 0];
    const float qy = ptsB[gq * 3 + 1];
    const float qz = ptsB[gq * 3 + 2];

    // B operand (4x16 f32): lanes 0-15 hold K=0,1 ; lanes 16-31 hold K=2,3
    // B col for query q = {qx, qy, qz, 1}
    v2f bq;
    bq.x = half ? qz   : qx;
    bq.y = half ? 1.0f : qy;

    float td[MAXK];
    int   ti[MAXK];
    #pragma unroll
    for (int j = 0; j < MAXK; ++j) { td[j] = 3.0e38f; ti[j] = 0; }

    for (int t = 0; t < N_ / 16; ++t) {
        const int cm = t * 16 + col;          // candidate this lane stages into A
        const float* cp = ptsB + cm * 3;
        __builtin_prefetch(cp + 48, 0, 0);    // next tile (speculative, global_prefetch_b8)
        const float cx = cp[0], cy = cp[1], cz = cp[2];
        const float cn = cx * cx + cy * cy + cz * cz;

        // A operand (16x4 f32): row m = candidate m; K = {-2x, -2y, -2z, |c|^2}
        // lanes 0-15: VGPR0=K0, VGPR1=K1 ; lanes 16-31: VGPR0=K2, VGPR1=K3
        v2f av;
        av.x = half ? (-2.0f * cz) : (-2.0f * cx);
        av.y = half ? cn           : (-2.0f * cy);

        v8f acc = {};
        // score[m][n] = |c_m|^2 - 2 c_m.q_n   (query norm dropped: rank-invariant)
        v8f s = __builtin_amdgcn_wmma_f32_16x16x4_f32(false, av, false, bq,
                                                      (short)0, acc, false, false);

        #pragma unroll
        for (int v = 0; v < 8; ++v) {
            float d = s[v];
            if (d < td[MAXK - 1]) {           // fast reject vs current worst
                int di = t * 16 + half * 8 + v;
                #pragma unroll
                for (int j = 0; j < MAXK; ++j) {
                    const bool sw = d < td[j];
                    const float od = td[j]; const int oi = ti[j];
                    td[j] = sw ? d  : od;
                    ti[j] = sw ? di : oi;
                    d  = sw ? od : d;
                    di = sw ? oi : di;
                }
            }
        }
    }

    // ---- merge the two sorted half-lists of the lane pair (shfl_xor 16) ----
    float pd[MAXK]; int pi[MAXK];
    #pragma unroll
    for (int j = 0; j < MAXK; ++j) {
        pd[j] = __shfl_xor(td[j], 16, 32);
        pi[j] = __shfl_xor(ti[j], 16, 32);
    }
    #pragma unroll
    for (int j = 0; j < MAXK; ++j) {
        float d = pd[j]; int di = pi[j];
        if (d < td[MAXK - 1]) {
            #pragma unroll
            for (int m = 0; m < MAXK; ++m) {
                const bool sw = d < td[m];
                const float od = td[m]; const int oi = ti[m];
                td[m] = sw ? d  : od;
                ti[m] = sw ? di : oi;
                d  = sw ? od : d;
                di = sw ? oi : di;
            }
        }
    }

    if (half == 0) {
        const int ql = wave * 16 + col;
        #pragma unroll
        for (int j = 0; j < MAXK; ++j) s_idx[ql][j] = ti[j];
    }
    __syncthreads();

    // ---- Phase 2: gather + mean over first k neighbors (L2-resident preds) ----
    const int q   = tid >> 1;                 // 0..127 (query within block)
    const int ch0 = (tid & 1) * 32;           // channel half
    const float* predsB = preds + (size_t)b * N_ * C_;

    float4 a[8];
    #pragma unroll
    for (int c = 0; c < 8; ++c) a[c] = make_float4(0.f, 0.f, 0.f, 0.f);

    for (int j = 0; j < kk; ++j) {
        const int id = s_idx[q][j];
        const float4* row = (const float4*)(predsB + (size_t)id * C_ + ch0);
        #pragma unroll
        for (int c = 0; c < 8; ++c) {
            const float4 v = row[c];
            a[c].x += v.x; a[c].y += v.y; a[c].z += v.z; a[c].w += v.w;
        }
    }
    const float inv = 1.0f / (float)kk;
    float4* op = (float4*)(out + ((size_t)b * N_ + (size_t)(qbase + q)) * C_ + ch0);
    #pragma unroll
    for (int c = 0; c < 8; ++c)
        op[c] = make_float4(a[c].x * inv, a[c].y * inv, a[c].z * inv, a[c].w * inv);
}

extern "C" void kernel_launch(void* const* d_in, const int* in_sizes, int n_in,
                              void* d_out, int out_size, void* d_ws, size_t ws_size,
                              hipStream_t stream) {
    const float* points = (const float*)d_in[0];
    const float* preds  = (const float*)d_in[1];
    const float* kvec   = (const float*)d_in[2];
    float* out = (float*)d_out;

    dim3 grid(B_ * (N_ / QPB));   // 4 * 32 = 128 workgroups
    dim3 block(TPB);              // 256 threads = 8 wave32
    hipLaunchKernelGGL(knn_space_mean_kernel, grid, block, 0, stream,
                       points, preds, kvec, out);
}